// Concat_84653805404632
// MI455X (gfx1250) — compile-verified
//
#include <hip/hip_runtime.h>
#include <hip/hip_bf16.h>

// Problem constants (match reference):
//   N_SUPPORT=10, N_QUERY=25, N_CLASS=2 -> ns_all=20, nq_all=50, pairs=1000
//   DIM0=128, FEAT=512 -> output (1000, 128, 1024) fp32 = 512 MB (store-bound)
#define NS_ALL 20
#define NQ_ALL 50
#define DIM0   128
#define FEAT   512
#define ROWS_PER_BLOCK 8      // D-rows handled per block
#define THREADS 256           // 8 wave32s
#define F4_PER_SRC_ROW (FEAT / 4)       // 128 float4 per 512-float source row
#define F4_PER_DST_ROW (2 * FEAT / 4)   // 256 float4 per 1024-float output row

__global__ __launch_bounds__(THREADS)
void bcast_concat_async_kernel(const float* __restrict__ x, float* __restrict__ out) {
    // 32 KB LDS staging buffer: ROWS_PER_BLOCK rows x 256 float4
    __shared__ float4 smem[ROWS_PER_BLOCK * 256];

    const unsigned t     = threadIdx.x;
    const unsigned b     = blockIdx.x;
    const unsigned pair  = b >> 4;                    // 16 blocks per (q,s) pair
    const unsigned dbase = (b & 15u) * ROWS_PER_BLOCK;
    const unsigned s     = pair % NS_ALL;
    const unsigned q     = pair / NS_ALL;

    // Source: threads 0..127 read the support row (first 512 floats of the
    // output row), threads 128..255 read the query row (second 512 floats).
    const float4* x4 = (const float4*)x;
    const float4* src4;
    unsigned col;
    if (t < 128u) {
        src4 = x4 + (size_t)(s * DIM0 + dbase) * F4_PER_SRC_ROW;
        col  = t;
    } else {
        src4 = x4 + (size_t)((NS_ALL + q) * DIM0 + dbase) * F4_PER_SRC_ROW;
        col  = t - 128u;
    }

    // LDS byte address of this thread's slot (generic->local is a 32-bit trunc
    // on amdgcn, so the low 32 bits of the flat address ARE the LDS address).
    const unsigned lds_base = (unsigned)(uintptr_t)(&smem[t]);

    // Stage global -> LDS with the CDNA5 async data mover (16 B per lane).
    // Each wave writes only its own LDS slice, so only the wave-private
    // ASYNCcnt wait is needed (no workgroup barrier).
#pragma unroll
    for (int i = 0; i < ROWS_PER_BLOCK; ++i) {
        const float4* g = src4 + (size_t)i * F4_PER_SRC_ROW + col;
        unsigned l = lds_base + (unsigned)(i * 256 * sizeof(float4));
        asm volatile("global_load_async_to_lds_b128 %0, %1, off"
                     :: "v"(l), "v"(g)
                     : "memory");
    }

    asm volatile("s_wait_asynccnt 0" ::: "memory");

    // Stream LDS -> global with non-temporal hint: the 512 MB output is never
    // re-read, and NT keeps the 18 MB (L2-resident) input hot.
    float4* dst4 = (float4*)out + (size_t)(pair * DIM0 + dbase) * F4_PER_DST_ROW + t;
#pragma unroll
    for (int i = 0; i < ROWS_PER_BLOCK; ++i) {
        float4* g = dst4 + (size_t)i * F4_PER_DST_ROW;
        unsigned l = lds_base + (unsigned)(i * 256 * sizeof(float4));
        asm volatile("global_store_async_from_lds_b128 %0, %1, off th:TH_STORE_NT"
                     :: "v"(g), "v"(l)
                     : "memory");
    }

    // Drain outstanding async stores before the wave retires (S_ENDPGM would
    // also imply a wait-idle, but be explicit).
    asm volatile("s_wait_asynccnt 0" ::: "memory");
}

extern "C" void kernel_launch(void* const* d_in, const int* in_sizes, int n_in,
                              void* d_out, int out_size, void* d_ws, size_t ws_size,
                              hipStream_t stream) {
    (void)in_sizes; (void)n_in; (void)d_ws; (void)ws_size; (void)out_size;
    const float* x = (const float*)d_in[0];   // (70, 128, 512) fp32
    float* out     = (float*)d_out;           // (1000, 128, 1024) fp32

    const int pairs = NQ_ALL * NS_ALL;                    // 1000
    const int blocks_per_pair = DIM0 / ROWS_PER_BLOCK;    // 16
    dim3 grid(pairs * blocks_per_pair);                   // 16000 blocks
    dim3 block(THREADS);
    bcast_concat_async_kernel<<<grid, block, 0, stream>>>(x, out);
}